// GO_sim_embedding_9457517986562
// MI455X (gfx1250) — compile-verified
//
#include <hip/hip_runtime.h>

// ---------------------------------------------------------------------------
// CDNA5 (gfx1250) GCN layer: xw = x@W via V_WMMA_F32_16X16X4_F32,
// edge scatter via f32 global atomics (L2-resident xw), fused bias/relu/resid.
// ---------------------------------------------------------------------------

typedef __attribute__((ext_vector_type(2))) float v2f;
typedef __attribute__((ext_vector_type(8))) float v8f;

#define N_HID 128

// out[n][f] = b[f] for all rows (accumulation target for the edge scatter)
__global__ void __launch_bounds__(256)
init_bias_kernel(float* __restrict__ out, const float* __restrict__ bias, int total) {
    int i = blockIdx.x * 256 + threadIdx.x;
    if (i < total) out[i] = bias[i & (N_HID - 1)];
}

// XW[M,128] = X[M,128] @ W[128,128], fp32, exact via V_WMMA_F32_16X16X4_F32.
// One wave computes one 16x16 tile of XW; 8 waves/block cover the 128 cols.
// grid.x = ceil(M/16), block = 256 (8 waves).
//
// ISA VGPR layouts (CDNA5 ISA, 32-bit A 16x4 / C-D 16x16):
//   A (16x4):  lanes 0-15 -> M=lane, {VGPR0,VGPR1} = K{0,1};
//              lanes 16-31 -> M=lane-16, {VGPR0,VGPR1} = K{2,3}
//   B (4x16):  lanes 0-15 -> N=lane, K{0,1}; lanes 16-31 -> N=lane-16, K{2,3}
//   C/D:       VGPR v, lanes 0-15 -> (M=v,   N=lane);
//                       lanes 16-31 -> (M=v+8, N=lane-16)
__global__ void __launch_bounds__(256)
gemm_wmma_f32_kernel(const float* __restrict__ X, const float* __restrict__ W,
                     float* __restrict__ XW, int M) {
    const int wave = threadIdx.x >> 5;   // 0..7 -> output col tile
    const int lane = threadIdx.x & 31;
    const int half = lane >> 4;          // 0 or 1
    const int l16  = lane & 15;
    const int m0   = blockIdx.x * 16;
    const int n0   = wave * 16;

    // Clamp A-row for tail tiles: WMMA needs EXEC all-1s, so loads stay
    // unconditional; garbage rows are simply not stored below.
    int arow = m0 + l16;
    if (arow >= M) arow = M - 1;
    const float* __restrict__ xrow = X + (size_t)arow * N_HID;
    const float* __restrict__ wcol = W + n0 + l16;

    v8f acc = {};
    #pragma unroll 4
    for (int kb = 0; kb < N_HID; kb += 4) {
        const int ka = kb + half * 2;    // this lane-half's K pair
        v2f a, b;
        a.x = xrow[ka];
        a.y = xrow[ka + 1];
        b.x = wcol[(size_t)ka * N_HID];
        b.y = wcol[(size_t)(ka + 1) * N_HID];
        acc = __builtin_amdgcn_wmma_f32_16x16x4_f32(
                /*neg_a=*/false, a, /*neg_b=*/false, b,
                /*c_mod=*/(short)0, acc, /*reuse_a=*/false, /*reuse_b=*/false);
    }

    // Tile-uniform guard hoisted out of the per-element stores: full tiles
    // (all but at most the last block) take an unconditional, clause-friendly
    // path of 8 constant-offset global_store_b32 from one base address.
    float* __restrict__ orow = XW + (size_t)(m0 + half * 8) * N_HID + n0 + l16;
    if (m0 + 16 <= M) {
        #pragma unroll
        for (int v = 0; v < 8; ++v)
            orow[(size_t)v * N_HID] = acc[v];
    } else {
        #pragma unroll
        for (int v = 0; v < 8; ++v)
            if (m0 + half * 8 + v < M)
                orow[(size_t)v * N_HID] = acc[v];
    }
}

// One wave per edge: 32 lanes x float4 = one coalesced 512B row of xw[src],
// scatter-added into out[dst] with f32 global atomics (resolved in L2).
// block = 256 threads = 8 edges.
__global__ void __launch_bounds__(256)
scatter_edges_kernel(const float* __restrict__ XW, const int* __restrict__ src,
                     const int* __restrict__ dst, float* __restrict__ out, int E) {
    const int e = blockIdx.x * 8 + (threadIdx.x >> 5);
    if (e >= E) return;
    const int lane = threadIdx.x & 31;

    const int s = src[e];
    const int d = dst[e];
    const float4 v = ((const float4*)(XW + (size_t)s * N_HID))[lane];
    float* orow = out + (size_t)d * N_HID + lane * 4;
    atomicAdd(orow + 0, v.x);
    atomicAdd(orow + 1, v.y);
    atomicAdd(orow + 2, v.z);
    atomicAdd(orow + 3, v.w);
}

// out = relu(out) + x   (bias already folded in via init)
__global__ void __launch_bounds__(256)
finalize_kernel(const float* __restrict__ X, float* __restrict__ out, int total) {
    int i = blockIdx.x * 256 + threadIdx.x;
    if (i < total) {
        float a = out[i];
        out[i] = fmaxf(a, 0.0f) + X[i];
    }
}

extern "C" void kernel_launch(void* const* d_in, const int* in_sizes, int n_in,
                              void* d_out, int out_size, void* d_ws, size_t ws_size,
                              hipStream_t stream) {
    // Input order (setup_inputs): h_mf, h_bp, h_cc, mf_src, mf_dst, bp_src,
    // bp_dst, cc_src, cc_dst, W_mf, b_mf, W_bp, b_bp, W_cc, b_cc
    const float* h[3]   = { (const float*)d_in[0], (const float*)d_in[1], (const float*)d_in[2] };
    const int*   src[3] = { (const int*)d_in[3], (const int*)d_in[5], (const int*)d_in[7] };
    const int*   dst[3] = { (const int*)d_in[4], (const int*)d_in[6], (const int*)d_in[8] };
    const float* Wm[3]  = { (const float*)d_in[9],  (const float*)d_in[11], (const float*)d_in[13] };
    const float* bm[3]  = { (const float*)d_in[10], (const float*)d_in[12], (const float*)d_in[14] };
    const int    Nn[3]  = { in_sizes[0] / N_HID, in_sizes[1] / N_HID, in_sizes[2] / N_HID };
    const int    Ee[3]  = { in_sizes[3], in_sizes[5], in_sizes[7] };

    float* xw = (float*)d_ws;               // reused per graph; <= 50000*128 f32
    float* out = (float*)d_out;

    size_t out_off = 0;
    for (int g = 0; g < 3; ++g) {
        const int N = Nn[g];
        const int E = Ee[g];
        const int total = N * N_HID;
        float* og = out + out_off;

        init_bias_kernel<<<(total + 255) / 256, 256, 0, stream>>>(og, bm[g], total);

        gemm_wmma_f32_kernel<<<(N + 15) / 16, 256, 0, stream>>>(h[g], Wm[g], xw, N);

        scatter_edges_kernel<<<(E + 7) / 8, 256, 0, stream>>>(xw, src[g], dst[g], og, E);

        finalize_kernel<<<(total + 255) / 256, 256, 0, stream>>>(h[g], og, total);

        out_off += (size_t)total;
    }
}